// TGMandTRM_25022479466834
// MI455X (gfx1250) — compile-verified
//
#include <hip/hip_runtime.h>
#include <math.h>

// dims
#define BB 8
#define CC 128
#define HH 256
#define RR 16

typedef float v2f __attribute__((ext_vector_type(2)));
typedef float v8f __attribute__((ext_vector_type(8)));

// ---------------------------------------------------------------------------
// Pass 1: per (b,c) slab: column sums over i -> hpart[b,c,j]; total -> c_raw[b,c]
// grid = B*C = 1024 blocks, 256 threads; fully coalesced 1KB row reads
// ---------------------------------------------------------------------------
__global__ __launch_bounds__(256) void pool1_kernel(const float* __restrict__ x,
                                                    float* __restrict__ c_raw,
                                                    float* __restrict__ hpart) {
    const int bc  = blockIdx.x;                 // b*128 + c
    const int tid = threadIdx.x;                // column j
    const float* base = x + (size_t)bc * (HH * HH);
    float acc = 0.f;
    #pragma unroll 4
    for (int i = 0; i < HH; ++i)
        acc += base[i * HH + tid];
    hpart[(size_t)bc * HH + tid] = acc;         // sum over i for column j

    __shared__ float red[256];
    red[tid] = acc;
    __syncthreads();
    for (int s = 128; s > 0; s >>= 1) {
        if (tid < s) red[tid] += red[tid + s];
        __syncthreads();
    }
    if (tid == 0) c_raw[bc] = red[0];           // sum over (i,j)
}

// ---------------------------------------------------------------------------
// Pass 2: per (b,i): sum over (c,j) -> w_raw[b,i]
// grid = B*H = 2048 blocks, 256 threads; coalesced row reads
// ---------------------------------------------------------------------------
__global__ __launch_bounds__(256) void pool2_kernel(const float* __restrict__ x,
                                                    float* __restrict__ w_raw) {
    const int b   = blockIdx.x >> 8;
    const int i   = blockIdx.x & 255;
    const int tid = threadIdx.x;                // column j
    const float* base = x + ((size_t)b * CC) * (HH * HH) + (size_t)i * HH;
    float acc = 0.f;
    #pragma unroll 4
    for (int c = 0; c < CC; ++c)
        acc += base[(size_t)c * (HH * HH) + tid];

    __shared__ float red[256];
    red[tid] = acc;
    __syncthreads();
    for (int s = 128; s > 0; s >>= 1) {
        if (tid < s) red[tid] += red[tid + s];
        __syncthreads();
    }
    if (tid == 0) w_raw[blockIdx.x] = red[0];
}

// ---------------------------------------------------------------------------
// softmax over lam (16 elems) -> lam_s
// ---------------------------------------------------------------------------
__global__ void softmax_kernel(const float* __restrict__ lam, float* __restrict__ lam_s) {
    if (threadIdx.x == 0) {
        float m = lam[0];
        for (int r = 1; r < RR; ++r) m = fmaxf(m, lam[r]);
        float e[RR], s = 0.f;
        for (int r = 0; r < RR; ++r) { e[r] = expf(lam[r] - m); s += e[r]; }
        float inv = 1.f / s;
        for (int r = 0; r < RR; ++r) lam_s[r] = e[r] * inv;
    }
}

// ---------------------------------------------------------------------------
// Normalize pools: h_vec[b,j] = sum_c hpart / (C*H); w_vec, c_vecn
// grid = 8 blocks (per b), 256 threads
// ---------------------------------------------------------------------------
__global__ __launch_bounds__(256) void norm_kernel(const float* __restrict__ c_raw,
                                                   const float* __restrict__ hpart,
                                                   const float* __restrict__ w_raw,
                                                   float* __restrict__ c_vecn,
                                                   float* __restrict__ h_vec,
                                                   float* __restrict__ w_vec) {
    const int b = blockIdx.x, j = threadIdx.x;
    float hs = 0.f;
    for (int c = 0; c < CC; ++c)
        hs += hpart[((size_t)(b * CC + c)) * HH + j];
    h_vec[b * HH + j] = hs * (1.f / (CC * HH));
    w_vec[b * HH + j] = w_raw[b * HH + j] * (1.f / (CC * HH));
    if (j < CC) c_vecn[b * CC + j] = c_raw[b * CC + j] * (1.f / (HH * HH));
}

// ---------------------------------------------------------------------------
// Small GEMMs + sigmoid; lam folded into sC.
// 81920 threads = 320 blocks x 256
// ---------------------------------------------------------------------------
__global__ __launch_bounds__(256) void gates_kernel(const float* __restrict__ c_vecn,
                                                    const float* __restrict__ h_vec,
                                                    const float* __restrict__ w_vec,
                                                    const float* __restrict__ W1,
                                                    const float* __restrict__ W2,
                                                    const float* __restrict__ W3,
                                                    const float* __restrict__ lam_s,
                                                    float* __restrict__ sC,
                                                    float* __restrict__ sH,
                                                    float* __restrict__ sW) {
    const int gid = blockIdx.x * 256 + threadIdx.x;
    if (gid < RR * BB * CC) {                         // sC: [r][b][o<128]
        const int r = gid >> 10, b = (gid >> 7) & 7, o = gid & 127;
        const float* w = W1 + ((size_t)(r * CC + o)) * CC;
        const float* v = c_vecn + b * CC;
        float dot = 0.f;
        for (int k = 0; k < CC; ++k) dot += v[k] * w[k];
        sC[gid] = lam_s[r] / (1.f + expf(-dot));
    } else if (gid < RR * BB * CC + RR * BB * HH) {   // sH: [r][b][o<256]
        const int g = gid - RR * BB * CC;
        const int r = g >> 11, b = (g >> 8) & 7, o = g & 255;
        const float* w = W2 + ((size_t)(r * HH + o)) * HH;
        const float* v = h_vec + b * HH;
        float dot = 0.f;
        for (int k = 0; k < HH; ++k) dot += v[k] * w[k];
        sH[g] = 1.f / (1.f + expf(-dot));
    } else {                                          // sW
        const int g = gid - RR * BB * CC - RR * BB * HH;
        const int r = g >> 11, b = (g >> 8) & 7, o = g & 255;
        const float* w = W3 + ((size_t)(r * HH + o)) * HH;
        const float* v = w_vec + b * HH;
        float dot = 0.f;
        for (int k = 0; k < HH; ++k) dot += v[k] * w[k];
        sW[g] = 1.f / (1.f + expf(-dot));
    }
}

// ---------------------------------------------------------------------------
// Main fused kernel: per-wave GEMM t[(i),j] = G[i,r] * sW[r,j] via
// V_WMMA_F32_16X16X4_F32 (M=16 i-rows, N=16 j-cols, 4 K-steps of 4),
// fused with x load, copy-half store, relu(x*t) store.
// grid = 2048 blocks x 256 threads (8 waves); wave task = (b, c, i-tile)
// ---------------------------------------------------------------------------
#define LP 272   // LDS row pitch (floats): 256+16 -> hi/lo lane halves hit disjoint banks

__global__ __launch_bounds__(256) void recon_kernel(const float* __restrict__ x,
                                                    const float* __restrict__ sC,
                                                    const float* __restrict__ sH,
                                                    const float* __restrict__ sW,
                                                    float* __restrict__ out) {
    __shared__ float lC[RR * CC];       // sC[r][c] for this b  (lam pre-folded)
    __shared__ float lH[RR * LP];       // sH[r][i]
    __shared__ float lW[RR * LP];       // sW[r][j]

    const int b = blockIdx.x >> 8;      // all 8 waves of a block share b

    for (int idx = threadIdx.x; idx < RR * CC; idx += 256) {
        const int r = idx >> 7, c = idx & 127;
        lC[idx] = sC[((size_t)(r * BB + b)) * CC + c];
    }
    for (int idx = threadIdx.x; idx < RR * HH; idx += 256) {
        const int r = idx >> 8, j = idx & 255;
        lH[r * LP + j] = sH[((size_t)(r * BB + b)) * HH + j];
        lW[r * LP + j] = sW[((size_t)(r * BB + b)) * HH + j];
    }
    __syncthreads();

    const int wv   = threadIdx.x >> 5;
    const int lane = threadIdx.x & 31;
    const int rem  = ((blockIdx.x & 255) << 3) + wv;   // 0..2047
    const int c    = rem >> 4;                          // 0..127
    const int i0   = (rem & 15) << 4;                   // i-tile base
    const int hi   = lane >> 4;                         // lane half
    const int ml   = lane & 15;                         // row (A/M) or col (B/N)

    // A fragments: A[m,k] = sC'[k,c] * sH[k, i0+m]; K striped per ISA layout:
    // VGPR v, lanes0-15 -> K = 4k+v, lanes16-31 -> K = 4k+v+2
    v2f a[4];
    #pragma unroll
    for (int k = 0; k < 4; ++k) {
        const int r0 = k * 4 + 2 * hi;
        a[k].x = lC[r0 * CC + c]       * lH[r0 * LP + i0 + ml];
        a[k].y = lC[(r0 + 1) * CC + c] * lH[(r0 + 1) * LP + i0 + ml];
    }

    const float* xb   = x   + ((size_t)(b * CC + c)) * (HH * HH) + (size_t)i0 * HH;
    float*       outA = out + ((size_t)(b * 2 * CC + c)) * (HH * HH) + (size_t)i0 * HH;
    float*       outB = outA + (size_t)CC * (HH * HH);

    for (int n = 0; n < 16; ++n) {
        const int j0 = n << 4;

        // B fragments: B[k,j] = sW[k, j0+jl]; same K striping as A
        v2f bf[4];
        #pragma unroll
        for (int k = 0; k < 4; ++k) {
            const int r0 = k * 4 + 2 * hi;
            bf[k].x = lW[r0 * LP + j0 + ml];
            bf[k].y = lW[(r0 + 1) * LP + j0 + ml];
        }

        v8f acc = {};
        #pragma unroll
        for (int k = 0; k < 4; ++k)
            acc = __builtin_amdgcn_wmma_f32_16x16x4_f32(
                false, a[k], false, bf[k], (short)0, acc, false, false);

        // D layout: VGPR p -> rows i0+p (lanes0-15) / i0+p+8 (lanes16-31), col j0+ml
        #pragma unroll
        for (int p = 0; p < 8; ++p) {
            const int idx2 = (p + 8 * hi) * HH + j0 + ml;
            const float xv = xb[idx2];
            const float t  = acc[p];
            outA[idx2] = xv;
            const float pr = xv * t;
            outB[idx2] = pr > 0.f ? pr : 0.f;
        }
    }
}

// ---------------------------------------------------------------------------
extern "C" void kernel_launch(void* const* d_in, const int* in_sizes, int n_in,
                              void* d_out, int out_size, void* d_ws, size_t ws_size,
                              hipStream_t stream) {
    const float* x   = (const float*)d_in[0];
    const float* W1  = (const float*)d_in[1];
    const float* W2  = (const float*)d_in[2];
    const float* W3  = (const float*)d_in[3];
    const float* lam = (const float*)d_in[4];
    float* out = (float*)d_out;

    float* ws = (float*)d_ws;
    float* c_raw  = ws;                    // 1024
    float* w_raw  = c_raw + BB * CC;       // 2048
    float* hpart  = w_raw + BB * HH;       // 8*128*256 = 262144
    float* h_vec  = hpart + BB * CC * HH;  // 2048
    float* w_vec  = h_vec + BB * HH;       // 2048
    float* c_vecn = w_vec + BB * HH;       // 1024
    float* lam_s  = c_vecn + BB * CC;      // 16
    float* sC     = lam_s + RR;            // 16384
    float* sH     = sC + RR * BB * CC;     // 32768
    float* sW     = sH + RR * BB * HH;     // 32768

    pool1_kernel<<<BB * CC, 256, 0, stream>>>(x, c_raw, hpart);
    pool2_kernel<<<BB * HH, 256, 0, stream>>>(x, w_raw);
    softmax_kernel<<<1, 32, 0, stream>>>(lam, lam_s);
    norm_kernel<<<BB, 256, 0, stream>>>(c_raw, hpart, w_raw, c_vecn, h_vec, w_vec);
    gates_kernel<<<320, 256, 0, stream>>>(c_vecn, h_vec, w_vec, W1, W2, W3,
                                          lam_s, sC, sH, sW);
    recon_kernel<<<BB * 256, 256, 0, stream>>>(x, sC, sH, sW, out);
    (void)in_sizes; (void)n_in; (void)out_size; (void)ws_size;
}